// Self_Attention_SMC_6949257085594
// MI455X (gfx1250) — compile-verified
//
#include <hip/hip_runtime.h>
#include <hip/hip_bf16.h>

typedef float v2f __attribute__((ext_vector_type(2)));
typedef float v8f __attribute__((ext_vector_type(8)));

#define B_ 32
#define P_ 10
#define S_ 1024
#define D_ 256
#define WINDOW_ 200
#define NEG_INF_ (-1e9f)

// out[M x 256] = A[M x 256] @ W[256 x 256] + bias ; one wave per 16x16 tile.
// A fragment (16x4 f32): lanes 0-15 -> K=k0,k0+1 ; lanes 16-31 -> K=k0+2,k0+3
// B fragment (4x16 f32): symmetric (lane = column), C/D per ISA 16x16 f32 layout.
__global__ __launch_bounds__(32)
void proj_wmma_kernel(const float* __restrict__ A, const float* __restrict__ W,
                      const float* __restrict__ bias, float* __restrict__ out) {
    const int n0   = blockIdx.x * 16;
    const int m0   = blockIdx.y * 16;
    const int lane = threadIdx.x;   // 0..31 (one wave32)
    const int half = lane >> 4;     // 0 or 1
    const int l15  = lane & 15;

    const float* arow = A + (size_t)(m0 + l15) * D_;
    const float* wcol = W + n0 + l15;

    v8f c = {};
    #pragma unroll 8
    for (int k0 = 0; k0 < D_; k0 += 4) {
        v2f a, b;
        a.x = arow[k0 + 2 * half + 0];
        a.y = arow[k0 + 2 * half + 1];
        b.x = wcol[(size_t)(k0 + 2 * half + 0) * D_];
        b.y = wcol[(size_t)(k0 + 2 * half + 1) * D_];
        // D = A(16x4) x B(4x16) + C, fp32 WMMA (8-arg form)
        c = __builtin_amdgcn_wmma_f32_16x16x4_f32(
            /*neg_a=*/false, a, /*neg_b=*/false, b,
            /*c_mod=*/(short)0, c, /*reuse_a=*/false, /*reuse_b=*/false);
    }

    const float bv = bias[n0 + l15];
    #pragma unroll
    for (int r = 0; r < 8; ++r) {
        out[(size_t)(m0 + r + 8 * half) * D_ + n0 + l15] = c[r] + bv;
    }
}

// Scatter freshly projected k,v rows into the copied caches at position t.
__global__ __launch_bounds__(256)
void kv_update_kernel(const float* __restrict__ knew, const float* __restrict__ vnew,
                      const int* __restrict__ tptr,
                      float* __restrict__ outK, float* __restrict__ outV) {
    const int t  = *tptr;
    const int bp = blockIdx.x;    // 0..319
    const int d  = threadIdx.x;   // 0..255
    const size_t dst = ((size_t)bp * S_ + t) * D_ + d;
    outK[dst] = knew[(size_t)bp * D_ + d];
    outV[dst] = vnew[(size_t)bp * D_ + d];
}

// One 256-thread block per (b,p): logits -> softmax (LDS) -> attn @ V.
__global__ __launch_bounds__(256)
void attn_kernel(const float* __restrict__ Q, const float* __restrict__ Kc,
                 const float* __restrict__ Vc, const int* __restrict__ tptr,
                 float* __restrict__ attn_out, float* __restrict__ z_ws) {
    __shared__ float q_sh[D_];
    __shared__ float a_sh[S_];
    __shared__ float red[256];

    const int bp  = blockIdx.x;
    const int tid = threadIdx.x;
    const int t   = *tptr;
    const int lo  = (t > WINDOW_) ? (t - WINDOW_) : 0;

    q_sh[tid] = Q[(size_t)bp * D_ + tid];
    __syncthreads();

    const float scale = 0.0625f;  // 1/sqrt(256)
    float lmax = -3.0e38f;
    for (int j = tid; j < S_; j += 256) {
        float val = NEG_INF_;
        if (j >= lo && j <= t) {
            const float4* krow = (const float4*)(Kc + ((size_t)bp * S_ + j) * D_);
            const float4* qv4  = (const float4*)q_sh;
            float acc = 0.0f;
            #pragma unroll 8
            for (int k = 0; k < D_ / 4; ++k) {
                const float4 kv = krow[k];
                const float4 qv = qv4[k];
                acc += kv.x * qv.x + kv.y * qv.y + kv.z * qv.z + kv.w * qv.w;
            }
            val = acc * scale;
        }
        a_sh[j] = val;
        lmax = fmaxf(lmax, val);
    }

    // block-wide max
    red[tid] = lmax;
    __syncthreads();
    for (int s = 128; s > 0; s >>= 1) {
        if (tid < s) red[tid] = fmaxf(red[tid], red[tid + s]);
        __syncthreads();
    }
    const float m = red[0];
    __syncthreads();

    // exp + block-wide sum (masked lanes contribute exactly 0, matching fp32 exp(-1e9) underflow)
    float lsum = 0.0f;
    for (int j = tid; j < S_; j += 256) {
        const float e = (j >= lo && j <= t) ? __expf(a_sh[j] - m) : 0.0f;
        a_sh[j] = e;
        lsum += e;
    }
    red[tid] = lsum;
    __syncthreads();
    for (int s = 128; s > 0; s >>= 1) {
        if (tid < s) red[tid] += red[tid + s];
        __syncthreads();
    }
    const float inv = 1.0f / red[0];

    for (int j = tid; j < S_; j += 256) {
        const float w = a_sh[j] * inv;
        a_sh[j] = w;
        attn_out[(size_t)bp * S_ + j] = w;
    }
    __syncthreads();

    // z[d] = sum_j w[j] * V[j][d] ; coalesced across d = tid
    float acc = 0.0f;
    const float* vcol = Vc + (size_t)bp * S_ * D_ + tid;
    for (int j = lo; j <= t; ++j) acc += a_sh[j] * vcol[(size_t)j * D_];
    z_ws[(size_t)bp * D_ + tid] = acc;
}

extern "C" void kernel_launch(void* const* d_in, const int* in_sizes, int n_in,
                              void* d_out, int out_size, void* d_ws, size_t ws_size,
                              hipStream_t stream) {
    (void)in_sizes; (void)n_in; (void)out_size; (void)ws_size;

    const float* x   = (const float*)d_in[0];
    const float* Kin = (const float*)d_in[1];
    const float* Vin = (const float*)d_in[2];
    const int*   tp  = (const int*)d_in[3];
    const float* Wq  = (const float*)d_in[4];
    const float* bq  = (const float*)d_in[5];
    const float* Wk  = (const float*)d_in[6];
    const float* bk  = (const float*)d_in[7];
    const float* Wv  = (const float*)d_in[8];
    const float* bv  = (const float*)d_in[9];
    const float* Wz  = (const float*)d_in[10];
    const float* bz  = (const float*)d_in[11];

    const size_t nBP = (size_t)B_ * P_;       // 320
    const size_t zN  = nBP * D_;              // 81920
    const size_t kvN = nBP * S_ * D_;         // 83886080

    float* out_z    = (float*)d_out;          // (B,P,1,D)
    float* out_K    = out_z + zN;             // (B,P,S,D)
    float* out_V    = out_K + kvN;            // (B,P,S,D)
    float* out_attn = out_V + kvN;            // (B,P,1,S)

    float* wsQ = (float*)d_ws;                // 320KB each
    float* wsK = wsQ + zN;
    float* wsV = wsK + zN;
    float* wsZ = wsV + zN;

    // 1) Bulk KV passthrough — dominant cost (~1.34 GB HBM traffic, ~58 us @ 23.3 TB/s)
    hipMemcpyAsync(out_K, Kin, kvN * sizeof(float), hipMemcpyDeviceToDevice, stream);
    hipMemcpyAsync(out_V, Vin, kvN * sizeof(float), hipMemcpyDeviceToDevice, stream);

    // 2) q/k/v projections via fp32 WMMA (one wave per 16x16 tile)
    dim3 pgrid(D_ / 16, (unsigned)(nBP / 16));  // (16, 20)
    proj_wmma_kernel<<<pgrid, 32, 0, stream>>>(x, Wq, bq, wsQ);
    proj_wmma_kernel<<<pgrid, 32, 0, stream>>>(x, Wk, bk, wsK);
    proj_wmma_kernel<<<pgrid, 32, 0, stream>>>(x, Wv, bv, wsV);

    // 3) Write k,v at position `timestep` into the copied caches
    kv_update_kernel<<<(unsigned)nBP, 256, 0, stream>>>(wsK, wsV, tp, out_K, out_V);

    // 4) Windowed single-query attention (reads L2-resident K/V window)
    attn_kernel<<<(unsigned)nBP, 256, 0, stream>>>(wsQ, out_K, out_V, tp, out_attn, wsZ);

    // 5) Output projection z @ Wz + bz
    proj_wmma_kernel<<<pgrid, 32, 0, stream>>>(wsZ, Wz, bz, out_z);
}